// Yolov8_82557861363907
// MI455X (gfx1250) — compile-verified
//
#include <hip/hip_runtime.h>
#include <stdint.h>

#define NC      80
#define NA      8400
#define NB      32
#define NCH     84          // 4 box + 80 class channels
#define NSORT   16384       // next pow2 >= NA
#define NNMS    1000        // MAX_NMS
#define MAXDET  300
#define THREADS 1024
#define PIXF    412.0f
#define IOUMULF 2.22f
#define WHOFF   7680.0f     // MAX_WH
#define BOXDIVF 412.1635f
#define CONFDIVF 32767.0f

// -------------------------------------------------------------------------
// Kernel 0: zero the output (rows >= cnt must be 0).
// -------------------------------------------------------------------------
__global__ void yolo_zero_out(float* __restrict__ out, int n) {
    int i = blockIdx.x * 256 + threadIdx.x;
    if (i < n) out[i] = 0.0f;
}

// -------------------------------------------------------------------------
// Kernel 1: per-anchor class max/argmax -> packed sort key + class id.
// Vectorized: each thread owns 4 consecutive anchors, so every channel
// read is a 16B global_load_b128 (512B per wave32 request) -- this kernel
// carries ~86 MB and sets the roofline (~4 us at 23.3 TB/s).
// key = score<<14 | (8399 - anchor): descending u32 sort == top_k order
// (score desc, anchor asc on ties).
// -------------------------------------------------------------------------
__global__ void yolo_score_kernel(const float* __restrict__ pred,
                                  uint32_t* __restrict__ keys,
                                  int* __restrict__ jbuf) {
    int a4 = blockIdx.x * 256 + threadIdx.x;      // group of 4 anchors
    int b  = blockIdx.y;
    if (a4 >= NA / 4) return;

    const float4* p = (const float4*)(pred + (size_t)b * NCH * NA + (size_t)4 * NA) + a4;
    float4 v = p[0];
    float m0 = v.x, m1 = v.y, m2 = v.z, m3 = v.w;
    int   j0 = 0,  j1 = 0,  j2 = 0,  j3 = 0;
#pragma unroll 4
    for (int c = 1; c < NC; ++c) {
        v = p[(size_t)c * (NA / 4)];
        if (v.x > m0) { m0 = v.x; j0 = c; }       // strict '>' == first-max argmax
        if (v.y > m1) { m1 = v.y; j1 = c; }
        if (v.z > m2) { m2 = v.z; j2 = c; }
        if (v.w > m3) { m3 = v.w; j3 = c; }
    }
    int a0 = a4 * 4;
    uint4 kv;
    kv.x = ((uint32_t)(int)m0 << 14) | (uint32_t)(NA - 1 - (a0 + 0));
    kv.y = ((uint32_t)(int)m1 << 14) | (uint32_t)(NA - 1 - (a0 + 1));
    kv.z = ((uint32_t)(int)m2 << 14) | (uint32_t)(NA - 1 - (a0 + 2));
    kv.w = ((uint32_t)(int)m3 << 14) | (uint32_t)(NA - 1 - (a0 + 3));
    ((uint4*)(keys + (size_t)b * NA))[a4] = kv;
    int4 jv = { j0, j1, j2, j3 };
    ((int4*)(jbuf + (size_t)b * NA))[a4] = jv;
}

// -------------------------------------------------------------------------
// Kernel 2: one 1024-thread workgroup per batch.
//   - async global->LDS copy of 8400 keys (gfx1250 ASYNCcnt path)
//   - in-LDS bitonic sort (descending) of 16384 keys (64 KB LDS)
//   - gather top-1000 boxes, greedy quantized-IoU NMS, write rows
// -------------------------------------------------------------------------
__global__ void __launch_bounds__(THREADS)
yolo_nms_kernel(const float* __restrict__ pred,
                const uint32_t* __restrict__ keys,
                const int* __restrict__ jbuf,
                float* __restrict__ out) {
    __shared__ uint32_t sh[NSORT];            // 64 KB
    const int tid = threadIdx.x;
    const int b   = blockIdx.x;
    const uint32_t* kb = keys + (size_t)b * NA;

    // ---- stage keys into LDS via async copies (tracked by ASYNCcnt) ----
    for (int base = 0; base < NA; base += THREADS) {
        int idx = base + tid;
        if (idx < NA) {
            uint32_t ldsoff = (uint32_t)(size_t)&sh[idx];   // low 32b = LDS byte addr
            const uint32_t* g = kb + idx;
            asm volatile("global_load_async_to_lds_b32 %0, %1, off"
                         :: "v"(ldsoff), "v"(g) : "memory");
        }
    }
    // pad [NA, NSORT) with minimal keys (plain DS stores)
    for (int idx = NA + tid; idx < NSORT; idx += THREADS) sh[idx] = 0u;
    asm volatile("s_wait_asynccnt 0x0" ::: "memory");
    __syncthreads();

    // ---- bitonic sort, descending ----
    for (unsigned k = 2; k <= NSORT; k <<= 1) {
        for (unsigned j = k >> 1; j > 0; j >>= 1) {
            for (unsigned s = 0; s < NSORT; s += THREADS) {
                unsigned i   = s + tid;
                unsigned ixj = i ^ j;
                if (ixj > i) {
                    uint32_t va = sh[i], vb = sh[ixj];
                    bool swp = ((i & k) == 0) ? (va < vb) : (va > vb);
                    if (swp) { sh[i] = vb; sh[ixj] = va; }
                }
            }
            __syncthreads();
        }
    }

    // thread t owns sorted entry t (top-1000 matter)
    uint32_t mykey = (tid < NNMS) ? sh[tid] : 0u;
    __syncthreads();   // done reading keys; LDS is reused below

    // overlay NMS arrays on the sort buffer (20 KB + 33 ints of the 64 KB)
    float* ox1  = (float*)sh;
    float* oy1  = ox1 + THREADS;
    float* ox2  = oy1 + THREADS;
    float* oy2  = ox2 + THREADS;
    float* oar  = oy2 + THREADS;
    int*   imin = (int*)(oar + THREADS);      // [0..31] per-wave, [32] broadcast

    int   score = (int)(mykey >> 14);
    int   a     = (NA - 1) - (int)(mykey & 0x3FFFu);
    bool  live  = (tid < NNMS) && (score > 8192);     // conf > CONF_THRES

    const float* pb = pred + (size_t)b * NCH * NA;
    float cx = pb[a], cy = pb[NA + a], w = pb[2 * NA + a], h = pb[3 * NA + a];
    int   jc = jbuf[b * NA + a];

    float px1 = truncf(cx - w * 0.5f);
    float py1 = truncf(cy - h * 0.5f);
    float px2 = truncf(cx + w * 0.5f);
    float py2 = truncf(cy + h * 0.5f);
    float off = (float)jc * WHOFF;            // class-offset for NMS only
    float mx1 = px1 + off, my1 = py1 + off, mx2 = px2 + off, my2 = py2 + off;
    float mar = (px2 - px1 + PIXF) * (py2 - py1 + PIXF);   // offsets cancel

    ox1[tid] = mx1; oy1[tid] = my1; ox2[tid] = mx2; oy2[tid] = my2; oar[tid] = mar;
    __syncthreads();

    // ---- greedy NMS: pick first live in sorted order, suppress, repeat ----
    const int wv = tid >> 5;                  // wave32: 32 waves per block
    int cnt = 0;
    for (;;) {
        if (tid < 33) imin[tid] = 0x7FFFFFFF;
        __syncthreads();
        if (live) atomicMin(&imin[wv], tid);  // ds_min per-wave slot
        __syncthreads();
        if (tid == 0) {
            int best = 0x7FFFFFFF;
#pragma unroll
            for (int q = 0; q < 32; ++q) best = min(best, imin[q]);
            imin[32] = best;
        }
        __syncthreads();
        int i = imin[32];
        if (i == 0x7FFFFFFF || cnt >= MAXDET) break;

        float bx1 = ox1[i], by1 = oy1[i], bx2 = ox2[i], by2 = oy2[i], ba = oar[i];
        if (live) {
            float xx1 = fmaxf(bx1, mx1);
            float yy1 = fmaxf(by1, my1);
            float xx2 = fminf(bx2, mx2);
            float yy2 = fminf(by2, my2);
            float ww  = fmaxf(0.0f, xx2 - xx1 + PIXF);
            float hh  = fmaxf(0.0f, yy2 - yy1 + PIXF);
            float inter = ww * hh * IOUMULF;
            if (inter > ba + mar - inter) live = false;
        }
        if (tid == i) {                       // kept box: emit row cnt
            live = false;
            float* o = out + ((size_t)b * MAXDET + cnt) * 6;
            o[0] = px1 / BOXDIVF;
            o[1] = py1 / BOXDIVF;
            o[2] = px2 / BOXDIVF;
            o[3] = py2 / BOXDIVF;
            o[4] = (float)score / CONFDIVF;
            o[5] = (float)jc;
        }
        cnt++;
        __syncthreads();
    }
}

// -------------------------------------------------------------------------
extern "C" void kernel_launch(void* const* d_in, const int* in_sizes, int n_in,
                              void* d_out, int out_size, void* d_ws, size_t ws_size,
                              hipStream_t stream) {
    (void)in_sizes; (void)n_in; (void)ws_size;
    const float* pred = (const float*)d_in[0];
    float* out = (float*)d_out;

    uint32_t* keys = (uint32_t*)d_ws;                 // 32*8400 u32 = 1.05 MB
    int*      jbuf = (int*)(keys + (size_t)NB * NA);  // 32*8400 i32 = 1.05 MB

    yolo_zero_out<<<(out_size + 255) / 256, 256, 0, stream>>>(out, out_size);

    dim3 g1((NA / 4 + 255) / 256, NB);                // 2100 anchor-quads / batch
    yolo_score_kernel<<<g1, 256, 0, stream>>>(pred, keys, jbuf);

    yolo_nms_kernel<<<NB, THREADS, 0, stream>>>(pred, keys, jbuf, out);
}